// roihead_70489003261994
// MI455X (gfx1250) — compile-verified
//
#include <hip/hip_runtime.h>
#include <hip/hip_bf16.h>
#include <math.h>

// ---------------- problem constants ----------------
#define R_ROIS   128
#define CCH      512
#define HH       37
#define WW       37
#define OUTP     7
#define SCALE    0.0625f
#define D_IN     (CCH * OUTP * OUTP)   // 25088
#define HID      4096
#define NCLS     21
#define NLOC     (NCLS * 4)            // 84

typedef float v2f __attribute__((ext_vector_type(2)));
typedef float v8f __attribute__((ext_vector_type(8)));

// Activation layout ("paired"): act2[(k>>1)][r][k&1], i.e. the two consecutive
// k-values a WMMA lane needs sit in one 8-byte word -> B frag = one b64 load.
__device__ __forceinline__ size_t pair_idx(int k, int r) {
    return (size_t)(k >> 1) * (2 * R_ROIS) + r * 2 + (k & 1);
}

// ---------------------------------------------------------------------------
// Kernel 1: ROI max pool -> paired-transposed activations act2[K/2][128][2].
// Thread mapping is ROI-fastest so stores are coalesced.
// ---------------------------------------------------------------------------
__global__ void roi_pool_t_kernel(const float* __restrict__ x,
                                  const float* __restrict__ rois,
                                  const int*   __restrict__ roi_idx,
                                  float* __restrict__ pool2) {
    int gid = blockIdx.x * blockDim.x + threadIdx.x;
    const int total = R_ROIS * CCH * OUTP * OUTP;
    if (gid >= total) return;

    int r  = gid % R_ROIS;
    int k  = gid / R_ROIS;            // k = c*49 + ph*7 + pw
    int pw = k % OUTP;
    int ph = (k / OUTP) % OUTP;
    int c  = k / (OUTP * OUTP);

    float y1 = roundf(rois[r * 4 + 0] * SCALE);
    float x1 = roundf(rois[r * 4 + 1] * SCALE);
    float y2 = roundf(rois[r * 4 + 2] * SCALE);
    float x2 = roundf(rois[r * 4 + 3] * SCALE);
    float rh = fmaxf(y2 - y1 + 1.0f, 1.0f);
    float rw = fmaxf(x2 - x1 + 1.0f, 1.0f);

    int hs = (int)fminf(fmaxf(y1 + floorf((float)ph * rh / 7.0f),         0.0f), (float)HH);
    int he = (int)fminf(fmaxf(y1 + ceilf(((float)ph + 1.0f) * rh / 7.0f), 0.0f), (float)HH);
    int ws = (int)fminf(fmaxf(x1 + floorf((float)pw * rw / 7.0f),         0.0f), (float)WW);
    int we = (int)fminf(fmaxf(x1 + ceilf(((float)pw + 1.0f) * rw / 7.0f), 0.0f), (float)WW);

    const float* xp = x + ((size_t)roi_idx[r] * CCH + c) * (HH * WW);
    float m = -INFINITY;
    for (int h = hs; h < he; ++h) {
        const float* row = xp + h * WW;
        for (int w = ws; w < we; ++w) m = fmaxf(m, row[w]);
    }
    if (m == -INFINITY) m = 0.0f;     // empty bin -> 0 (matches reference)

    pool2[pair_idx(k, r)] = m;
}

// ---------------------------------------------------------------------------
// Kernel 2: fp32 WMMA GEMM   out = act( A[M][K] * B[K][128] + bias )
//   A  : weights row-major (streamed from HBM exactly once via LDS)
//   B2 : activations in paired layout [K/2][128][2] (hot in L2)
// 256 threads = 8 waves; block owns one 16-row M-tile (LDS-shared weights),
// wave w owns ROI tile w. Dual accumulators break the WMMA RAW chain.
// ---------------------------------------------------------------------------
#define KC  64
#define LDT 66   // padded LDS row stride (floats) to avoid bank conflicts

__global__ __launch_bounds__(256) void gemm_wmma_f32(
    const float* __restrict__ A,     // [M][K]
    const float* __restrict__ B2,    // [K/2][128][2]
    const float* __restrict__ bias,  // [M]
    float* __restrict__ out,
    int M, int K, int do_relu, int transposed_out, int ldo) {

    __shared__ float wtile[16 * LDT];

    const int lane = threadIdx.x & 31;
    const int wave = threadIdx.x >> 5;        // 0..7 -> ROI tile
    const int m0   = blockIdx.x * 16;
    const int r0   = wave * 16;
    const int mrow = lane & 15;               // M index inside fragment
    const int koff = (lane >> 4) * 2;         // lanes 16-31 hold K+2,K+3
    const int rr   = r0 + mrow;               // this lane's ROI column

    // cooperative A-tile load indexing: 16 rows x 64 floats, float4 per thread
    const int lrow = threadIdx.x >> 4;        // 0..15
    const int lkg  = (threadIdx.x & 15) * 4;  // 0,4,...,60
    int grow = m0 + lrow; if (grow >= M) grow = M - 1;   // clamp (writes guarded)
    const float* arow = A + (size_t)grow * K;

    v8f acc0 = {};
    v8f acc1 = {};

    for (int k0 = 0; k0 < K; k0 += KC) {
        // stage W tile in LDS (coalesced 256B rows)
        *(float4*)(&wtile[lrow * LDT + lkg]) = *(const float4*)(arow + k0 + lkg);
        __syncthreads();

        // prefetch next weight chunk while computing (global_prefetch_b8)
        if (k0 + KC < K)
            __builtin_prefetch(arow + k0 + KC + lkg, 0, 3);

        // base of this lane's B-pair stream: pair index (k0+koff)/2, column rr
        const v2f* bptr = (const v2f*)(B2 + (size_t)((k0 + koff) >> 1) * (2 * R_ROIS) + rr * 2);
        // consecutive WMMA steps advance B by one pair index (k by 2) -> 256 floats
        const int BSTEP = 2 * R_ROIS / 2;     // in v2f units: one pair index = 128 v2f

        #pragma unroll
        for (int kk = 0; kk < KC; kk += 8) {
            v2f a0 = *(const v2f*)(&wtile[mrow * LDT + kk + koff]);
            v2f b0 = bptr[0];
            acc0 = __builtin_amdgcn_wmma_f32_16x16x4_f32(
                false, a0, false, b0, (short)0, acc0, false, false);

            v2f a1 = *(const v2f*)(&wtile[mrow * LDT + kk + 4 + koff]);
            v2f b1 = bptr[BSTEP];
            acc1 = __builtin_amdgcn_wmma_f32_16x16x4_f32(
                false, a1, false, b1, (short)0, acc1, false, false);

            bptr += 2 * BSTEP;
        }
        __syncthreads();
    }

    // merge dual accumulators, then epilogue.
    // C/D layout: VGPR j -> lanes0-15 M=j, lanes16-31 M=j+8; N=lane&15
    const int col = rr;
    #pragma unroll
    for (int j = 0; j < 8; ++j) {
        int row = m0 + j + (lane >> 4) * 8;
        if (row < M) {
            float v = acc0[j] + acc1[j] + bias[row];
            if (do_relu) v = fmaxf(v, 0.0f);
            if (transposed_out) out[(size_t)col * ldo + row] = v;   // [R, M] final
            else                out[pair_idx(row, col)]      = v;   // paired actT
        }
    }
}

// ---------------------------------------------------------------------------
extern "C" void kernel_launch(void* const* d_in, const int* in_sizes, int n_in,
                              void* d_out, int out_size, void* d_ws, size_t ws_size,
                              hipStream_t stream) {
    // setup_inputs order:
    const float* x        = (const float*)d_in[0];
    const float* rois     = (const float*)d_in[1];
    const int*   roi_idx  = (const int*)  d_in[2];
    const float* W1       = (const float*)d_in[3];
    const float* b1       = (const float*)d_in[4];
    const float* W2       = (const float*)d_in[5];
    const float* b2       = (const float*)d_in[6];
    const float* W_loc    = (const float*)d_in[7];
    const float* b_loc    = (const float*)d_in[8];
    const float* W_score  = (const float*)d_in[9];
    const float* b_score  = (const float*)d_in[10];

    float* out = (float*)d_out;          // [128*21] scores then [128*84] locs

    // workspace layout (all in paired activation format)
    float* pool2 = (float*)d_ws;                          // [25088/2][128][2]
    float* fc6T  = pool2 + (size_t)D_IN * R_ROIS;         // [4096/2][128][2]
    float* fc7T  = fc6T  + (size_t)HID  * R_ROIS;         // [4096/2][128][2]

    // 1) ROI max pool (paired-transposed output)
    {
        int total  = R_ROIS * CCH * OUTP * OUTP;
        int blocks = (total + 255) / 256;
        roi_pool_t_kernel<<<blocks, 256, 0, stream>>>(x, rois, roi_idx, pool2);
    }

    // 2) fc6 = relu(pool @ W1^T + b1)
    gemm_wmma_f32<<<HID / 16, 256, 0, stream>>>(
        W1, pool2, b1, fc6T, HID, D_IN, /*relu*/1, /*tout*/0, 0);

    // 3) fc7 = relu(fc6 @ W2^T + b2)
    gemm_wmma_f32<<<HID / 16, 256, 0, stream>>>(
        W2, fc6T, b2, fc7T, HID, HID, /*relu*/1, /*tout*/0, 0);

    // 4) roi_score = fc7 @ W_score^T + b_score -> out[0 .. 128*21)
    gemm_wmma_f32<<<(NCLS + 15) / 16, 256, 0, stream>>>(
        W_score, fc7T, b_score, out, NCLS, HID, /*relu*/0, /*tout*/1, NCLS);

    // 5) roi_cls_loc = fc7 @ W_loc^T + b_loc -> out[128*21 .. )
    gemm_wmma_f32<<<(NLOC + 15) / 16, 256, 0, stream>>>(
        W_loc, fc7T, b_loc, out + (size_t)R_ROIS * NCLS, NLOC, HID,
        /*relu*/0, /*tout*/1, NLOC);
}